// AutonomyCost_89644557402528
// MI455X (gfx1250) — compile-verified
//
#include <hip/hip_runtime.h>
#include <hip/hip_bf16.h>

typedef __attribute__((ext_vector_type(2))) float v2f;
typedef __attribute__((ext_vector_type(8))) float v8f;
typedef __attribute__((ext_vector_type(2))) int   v2i;

#define EPSILON 1e-9f
#define N_QUAD 100

// ---------------------------------------------------------------------------
// Kernel 1: compute the row-independent scalar  scattering(t) + drag(t)
// into d_ws[0].  One wave, trivial cost.
// ---------------------------------------------------------------------------
__global__ void autonomy_scalar_kernel(const float* __restrict__ t_ptr,
                                       float* __restrict__ ws) {
    if (threadIdx.x == 0) {
        float t = t_ptr[0];
        float scattering = -0.5f * t * logf(t + EPSILON);
        // trapezoid rule for exp(-E) on linspace(0, t, 100)
        float h = t / (float)(N_QUAD - 1);
        float acc = 0.0f;
        for (int i = 0; i < N_QUAD; ++i) {
            float w = (i == 0 || i == N_QUAD - 1) ? 0.5f : 1.0f;
            acc += w * expf(-h * (float)i);
        }
        ws[0] = scattering + acc * h;
    }
}

// ---------------------------------------------------------------------------
// Kernel 2: gather + WMMA row-reduction.
// One wave handles a 16-row tile.  Per K-chunk of 4:
//   lane l :  row = base + (l & 15),  kpair = kb + 2*(l>>4)
//   B64 load of idx[row*L + kpair .. +1]  ->  two gathers from eta
//   -> exactly the V_WMMA_F32_16X16X4_F32 A-matrix layout
//   (VGPR0 = K even, VGPR1 = K odd; lanes 0-15 = K 0/1, lanes 16-31 = K 2/3)
// B = all-ones 4x16  =>  C[m][n] accumulates the row sum for every n.
// ---------------------------------------------------------------------------
__global__ __launch_bounds__(256) void autonomy_trace_kernel(
    const int*   __restrict__ idx,
    const float* __restrict__ eta,
    const float* __restrict__ ws,
    float*       __restrict__ out,
    int L, int rows) {

    const int lane        = threadIdx.x & 31;
    const int waveInBlock = threadIdx.x >> 5;
    const int wave        = blockIdx.x * (blockDim.x >> 5) + waveInBlock;
    const long rowBase    = (long)wave * 16;
    if (rowBase >= rows) return;   // wave-uniform: EXEC stays all-ones

    const int  r     = lane & 15;        // row within the 16-row tile
    const int  khalf = (lane >> 4) * 2;  // 0 for lanes 0-15, 2 for lanes 16-31

    const int* p = idx + (rowBase + r) * (long)L + khalf;

    v2f bones; bones.x = 1.0f; bones.y = 1.0f;   // all-ones B (layout-invariant)
    v8f c0 = {};
    v8f c1 = {};

    // 64 iterations for L=512; two accumulators break the WMMA D->C chain.
    #pragma unroll 4
    for (int kb = 0; kb < L; kb += 8) {
        v2i i0 = *(const v2i*)(p);
        v2i i1 = *(const v2i*)(p + 4);
        p += 8;

        v2f a0, a1;
        a0.x = eta[i0.x];
        a0.y = eta[i0.y];
        a1.x = eta[i1.x];
        a1.y = eta[i1.y];

        c0 = __builtin_amdgcn_wmma_f32_16x16x4_f32(
                 false, a0, false, bones, (short)0, c0, false, false);
        c1 = __builtin_amdgcn_wmma_f32_16x16x4_f32(
                 false, a1, false, bones, (short)0, c1, false, false);
    }

    const float scalar = ws[0];   // scattering + drag, uniform

    // C/D layout: row m (m<8)  -> VGPR m,   lanes 0-15
    //             row m (m>=8) -> VGPR m-8, lanes 16-31
    // All columns of C are equal (B == ones), so any owned lane works.
    const int j = lane & 7;
    float val = c0[0] + c1[0];
    val = (j == 1) ? (c0[1] + c1[1]) : val;
    val = (j == 2) ? (c0[2] + c1[2]) : val;
    val = (j == 3) ? (c0[3] + c1[3]) : val;
    val = (j == 4) ? (c0[4] + c1[4]) : val;
    val = (j == 5) ? (c0[5] + c1[5]) : val;
    val = (j == 6) ? (c0[6] + c1[6]) : val;
    val = (j == 7) ? (c0[7] + c1[7]) : val;

    if ((lane & 8) == 0) {                      // lanes 0-7 and 16-23 write
        const int m = ((lane & 16) ? 8 : 0) + j;
        out[rowBase + m] = val + scalar;
    }
}

// ---------------------------------------------------------------------------
// Host-side launcher
// ---------------------------------------------------------------------------
extern "C" void kernel_launch(void* const* d_in, const int* in_sizes, int n_in,
                              void* d_out, int out_size, void* d_ws, size_t ws_size,
                              hipStream_t stream) {
    const int*   idx = (const int*)  d_in[0];   // decision_indices [B, L]
    const float* eta = (const float*)d_in[1];   // eta_table [V]
    const float* t   = (const float*)d_in[2];   // scalar t
    float*       out = (float*)d_out;           // [B]
    float*       ws  = (float*)d_ws;

    const int rows = out_size;                  // B = 65536
    const int L    = in_sizes[0] / rows;        // 512

    autonomy_scalar_kernel<<<1, 32, 0, stream>>>(t, ws);

    const int wavesPerBlock = 8;                // 256 threads, wave32
    const int waves  = (rows + 15) / 16;        // 16 rows per wave
    const int blocks = (waves + wavesPerBlock - 1) / wavesPerBlock;

    autonomy_trace_kernel<<<blocks, wavesPerBlock * 32, 0, stream>>>(
        idx, eta, ws, out, L, rows);
}